// QAPLoss_41867341201849
// MI455X (gfx1250) — compile-verified
//
#include <hip/hip_runtime.h>
#include <math.h>

typedef __attribute__((ext_vector_type(2))) float v2f;
typedef __attribute__((ext_vector_type(8))) float v8f;

#define NBIN 25
#define DELTA_INV 12.0f          /* 1/DELTA, DELTA = 2/24 */
#define EPS_COS 1e-8f
#define EPS_P 1e-16f
#define TILES_PER_CHUNK 24       /* 16-row tiles per workgroup: 3 per wave */
#define WAVES_PER_WG 8

// Main streaming kernel: one pass over dXs.
// grid = (chunks, B), block = 256 (8 wave32 waves). Each WG owns batch b and
// tiles [chunk*24, chunk*24+24). Each wave processes a 16-row tile with
// V_WMMA_F32_16X16X4_F32, fusing the row sum-of-squares, then soft-bins the
// 16 sims into LDS accumulators. 51 partials per WG are written to d_ws.
__global__ __launch_bounds__(256) void qap_main_kernel(
    const float* __restrict__ qX, const float* __restrict__ dXs,
    const int* __restrict__ labels, float* __restrict__ partials,
    int n, int tilesPerB, int chunks)
{
    __shared__ float bins[51];   // [0..24]=num, [25..49]=den, [50]=labelSum
    const int tid  = threadIdx.x;
    const int lane = tid & 31;
    const int wave = tid >> 5;
    const int b     = blockIdx.y;
    const int chunk = blockIdx.x;

    if (tid < 51) bins[tid] = 0.0f;
    __syncthreads();

    const int h = lane >> 4;          // lane half: selects K phase {0,1} vs {2,3}
    const int r16 = lane & 15;        // A-matrix row index for this lane

    // ---- Preload B fragments (qX[b] broadcast along the 16 N columns) ----
    // B(4x16) layout mirrors A: VGPR0 = K=2h, VGPR1 = K=2h+1 per K-step.
    const float* q = qX + (size_t)b * 128;
    v2f qf[32];
    float qss = 0.0f;
#pragma unroll
    for (int c = 0; c < 32; ++c) {
        qf[c] = *(const v2f*)(q + 4 * c + 2 * h);
        qss = fmaf(qf[c].x, qf[c].x, qss);
        qss = fmaf(qf[c].y, qf[c].y, qss);
    }
    qss += __shfl_xor(qss, 16);                 // combine K-phase halves
    const float qn = fmaxf(sqrtf(qss), EPS_COS);

    for (int tt = wave; tt < TILES_PER_CHUNK; tt += WAVES_PER_WG) {
        const int tile = chunk * TILES_PER_CHUNK + tt;
        if (tile >= tilesPerB) continue;        // wave-uniform: EXEC stays full
        const int rowBase = tile * 16;
        int rowLoad = rowBase + r16;
        if (rowLoad > n - 1) rowLoad = n - 1;   // clamp (masked in binning)
        const float* Abase =
            dXs + ((size_t)b * n + rowLoad) * 128 + 2 * h;

        // ---- Stage the full tile: 32 simultaneously-live b64 loads ----
        // All a[c] are live at once -> allocator cannot recycle registers,
        // scheduler must keep 32 loads (8 KB/wave) in flight.
        v2f a[32];
#pragma unroll
        for (int c = 0; c < 32; ++c)
            a[c] = *(const v2f*)(Abase + 4 * c);     // global_load_b64

        // Prefetch next tile's rows (global_prefetch_b8) one tile ahead.
        if (tile + WAVES_PER_WG < tilesPerB)
            __builtin_prefetch(Abase + (size_t)WAVES_PER_WG * 16 * 128, 0, 1);

        v8f acc = {};                           // 16x16 f32 accumulator
        float s0 = 0.0f, s1 = 0.0f, s2 = 0.0f, s3 = 0.0f;
#pragma unroll
        for (int c = 0; c < 32; c += 4) {
            s0 = fmaf(a[c + 0].x, a[c + 0].x, s0);
            s0 = fmaf(a[c + 0].y, a[c + 0].y, s0);
            s1 = fmaf(a[c + 1].x, a[c + 1].x, s1);
            s1 = fmaf(a[c + 1].y, a[c + 1].y, s1);
            s2 = fmaf(a[c + 2].x, a[c + 2].x, s2);
            s2 = fmaf(a[c + 2].y, a[c + 2].y, s2);
            s3 = fmaf(a[c + 3].x, a[c + 3].x, s3);
            s3 = fmaf(a[c + 3].y, a[c + 3].y, s3);
            // D = A(16x4) * B(4x16) + C ; B columns replicated -> every
            // column of D holds the 16 row-dot-products.
            acc = __builtin_amdgcn_wmma_f32_16x16x4_f32(
                false, a[c + 0], false, qf[c + 0], (short)0, acc, false, false);
            acc = __builtin_amdgcn_wmma_f32_16x16x4_f32(
                false, a[c + 1], false, qf[c + 1], (short)0, acc, false, false);
            acc = __builtin_amdgcn_wmma_f32_16x16x4_f32(
                false, a[c + 2], false, qf[c + 2], (short)0, acc, false, false);
            acc = __builtin_amdgcn_wmma_f32_16x16x4_f32(
                false, a[c + 3], false, qf[c + 3], (short)0, acc, false, false);
        }
        float ssq = (s0 + s1) + (s2 + s3);

        // Full row sum-of-squares: lane r + lane r+16 hold the two K phases.
        ssq += __shfl_xor(ssq, 16);
        // Lane l (with (l&15)<8) owns row r = 8*h + (l&7); fetch its norm.
        const int r = (h << 3) | (lane & 7);
        const float ssr = __shfl(ssq, r);       // ds_bpermute

        // Extract this lane's dot product: acc element (M = (l&7)+8h).
        const int idx = lane & 7;
        float dot = acc[0];
        dot = (idx == 1) ? acc[1] : dot;
        dot = (idx == 2) ? acc[2] : dot;
        dot = (idx == 3) ? acc[3] : dot;
        dot = (idx == 4) ? acc[4] : dot;
        dot = (idx == 5) ? acc[5] : dot;
        dot = (idx == 6) ? acc[6] : dot;
        dot = (idx == 7) ? acc[7] : dot;

        const int row = rowBase + r;
        if (((lane & 15) < 8) && (row < n)) {
            const float lab = (float)labels[(size_t)b * n + row];
            const float dn = fmaxf(sqrtf(ssr), EPS_COS);
            const float sim = dot / (qn * dn);
            // Triangular soft-assignment: x = (1-sim)/DELTA hits bins
            // floor(x), floor(x)+1 with weights (1-f), f.
            const float x = (1.0f - sim) * DELTA_INV;
            const float mf = floorf(x);
            const int m = (int)mf;
            const float f = x - mf;
            const float w0 = 1.0f - f;
            if (m >= 0 && m < NBIN) {
                atomicAdd(&bins[m], w0 * lab);
                atomicAdd(&bins[NBIN + m], w0);
            }
            if (m + 1 >= 0 && m + 1 < NBIN) {
                atomicAdd(&bins[m + 1], f * lab);
                atomicAdd(&bins[NBIN + m + 1], f);
            }
            atomicAdd(&bins[50], lab);
        }
    }

    __syncthreads();
    if (tid < 51)
        partials[((size_t)b * chunks + chunk) * 51 + tid] = bins[tid];
}

// Tiny epilogue: reduce per-WG partials, run the 25-bin cumsum AP, mean.
__global__ __launch_bounds__(256) void qap_finish_kernel(
    const float* __restrict__ partials, int chunks, int Bc,
    float* __restrict__ out)
{
    __shared__ float acc[16 * 51];
    __shared__ float apLds[16];
    const int tid = threadIdx.x;

    for (int s = tid; s < Bc * 51; s += blockDim.x) {
        const int b = s / 51, i = s % 51;
        const float* p = partials + ((size_t)b * chunks) * 51 + i;
        float sum = 0.0f;
        for (int c = 0; c < chunks; ++c) sum += p[(size_t)c * 51];
        acc[s] = sum;
    }
    __syncthreads();

    if (tid < Bc) {
        const float* nb = &acc[tid * 51];
        const float* db = nb + NBIN;
        const float labSum = nb[50];
        float cnum = 0.0f, cden = EPS_P, ap = 0.0f;
        for (int k = 0; k < NBIN; ++k) {
            cnum += nb[k];
            cden += db[k];
            ap += (cnum / cden) * (nb[k] / labSum);
        }
        apLds[tid] = ap;
    }
    __syncthreads();

    if (tid == 0) {
        float s = 0.0f;
        for (int b = 0; b < Bc; ++b) s += apLds[b];
        out[0] = s / (float)Bc;
    }
}

extern "C" void kernel_launch(void* const* d_in, const int* in_sizes, int n_in,
                              void* d_out, int out_size, void* d_ws, size_t ws_size,
                              hipStream_t stream) {
    const float* qX     = (const float*)d_in[0];
    const float* dXs    = (const float*)d_in[1];
    const int*   labels = (const int*)d_in[2];
    float* out = (float*)d_out;

    const int Bc = in_sizes[0] / 128;                 // 16
    const int N  = in_sizes[1] / (Bc * 128);          // 50000
    const int tilesPerB = (N + 15) / 16;              // 3125
    const int chunks = (tilesPerB + TILES_PER_CHUNK - 1) / TILES_PER_CHUNK; // 131

    float* partials = (float*)d_ws;                   // B*chunks*51 floats (~428 KB)

    dim3 grid(chunks, Bc);
    qap_main_kernel<<<grid, 256, 0, stream>>>(qX, dXs, labels, partials,
                                              N, tilesPerB, chunks);
    qap_finish_kernel<<<1, 256, 0, stream>>>(partials, chunks, Bc, out);
}